// AttentionGAT_16338055594036
// MI455X (gfx1250) — compile-verified
//
#include <hip/hip_runtime.h>
#include <math.h>

typedef __attribute__((ext_vector_type(2))) float v2f;
typedef __attribute__((ext_vector_type(8))) float v8f;

// ---------------- helpers ----------------

__device__ __forceinline__ int enc_f32(float f) {
    int i = __float_as_int(f);
    return i >= 0 ? i : (i ^ 0x7fffffff);   // monotone float->int mapping
}
__device__ __forceinline__ float dec_f32(int i) {
    return __int_as_float(i >= 0 ? i : (i ^ 0x7fffffff));
}
__device__ __forceinline__ float lrelu02(float v) {
    return v > 0.f ? v : 0.2f * v;
}

// ---------------- kernels ----------------

// H[n,64] = X[n,64] @ W[64,64], fp32 exact via V_WMMA_F32_16X16X4_F32.
// One wave per 16x16 output tile; K-loop of 16 steps of K=4.
__global__ void gat_gemm_n64(const float* __restrict__ X, const float* __restrict__ W,
                             float* __restrict__ H, int n) {
    int wave = (blockIdx.x * blockDim.x + threadIdx.x) >> 5;
    int lane = threadIdx.x & 31;
    int tiles_m = n >> 4;                 // n is a multiple of 16 (50000 = 3125*16)
    int total = tiles_m * 4;              // 4 column tiles (64/16)
    if (wave >= total) return;
    int tm = wave >> 2;
    int tn = wave & 3;

    int m15   = lane & 15;
    int khalf = (lane >> 4) << 1;         // 0 for lanes 0-15, 2 for lanes 16-31
    int row   = tm * 16 + m15;
    int col   = tn * 16 + m15;

    const float2* xrow = (const float2*)(X + row * 64);

    v8f c = {};
    #pragma unroll
    for (int k = 0; k < 64; k += 4) {
        // A fragment: 16x4 fp32. lane<16 -> K=k+0,k+1 ; lane>=16 -> K=k+2,k+3
        float2 xa = xrow[(k + khalf) >> 1];
        v2f a; a.x = xa.x; a.y = xa.y;
        // B fragment: 4x16 fp32. lane<16 -> rows k+0,k+1 ; lane>=16 -> rows k+2,k+3
        v2f b;
        b.x = W[(k + khalf) * 64 + col];
        b.y = W[(k + khalf + 1) * 64 + col];
        c = __builtin_amdgcn_wmma_f32_16x16x4_f32(false, a, false, b,
                                                  (short)0, c, false, false);
    }
    // C/D layout: lanes 0-15: VGPR i -> M=i ; lanes 16-31: VGPR i -> M=8+i ; N = lane&15
    int rowBase = tm * 16 + ((lane >> 4) << 3);
    float* Hp = H + rowBase * 64 + col;
    #pragma unroll
    for (int i = 0; i < 8; ++i) Hp[i * 64] = c[i];
}

// es[i] = dot(H[i], a_src), ed[i] = dot(H[i], a_dst)
__global__ void gat_node_dots(const float* __restrict__ H,
                              const float* __restrict__ asrc, const float* __restrict__ adst,
                              float* __restrict__ es, float* __restrict__ ed, int n) {
    int i = blockIdx.x * blockDim.x + threadIdx.x;
    if (i >= n) return;
    const float4* h4 = (const float4*)(H + i * 64);
    const float4* a4 = (const float4*)asrc;
    const float4* b4 = (const float4*)adst;
    float s = 0.f, d = 0.f;
    #pragma unroll
    for (int j = 0; j < 16; ++j) {
        float4 h = h4[j], a = a4[j], b = b4[j];
        s += h.x*a.x + h.y*a.y + h.z*a.z + h.w*a.w;
        d += h.x*b.x + h.y*b.y + h.z*b.z + h.w*b.w;
    }
    es[i] = s; ed[i] = d;
}

__global__ void gat_init_nd(int* __restrict__ mmax, float* __restrict__ denom, int n) {
    int i = blockIdx.x * blockDim.x + threadIdx.x;
    if (i >= n) return;
    mmax[i] = enc_f32(-INFINITY);
    denom[i] = 0.f;
}

__global__ void gat_zero_f32(float* __restrict__ p, int count) {
    int i = blockIdx.x * blockDim.x + threadIdx.x;
    if (i < count) p[i] = 0.f;
}

// Pass 1: per-dst running max of leaky_relu(es[src]+ed[dst]) incl. self loops
__global__ void gat_edge_max(const int* __restrict__ eidx, int E, int n,
                             const float* __restrict__ es, const float* __restrict__ ed,
                             int* __restrict__ mmax) {
    int i = blockIdx.x * blockDim.x + threadIdx.x;
    int total = E + n;
    if (i >= total) return;
    int s, d;
    if (i < E) { s = eidx[i]; d = eidx[E + i]; } else { s = d = i - E; }
    float e = lrelu02(es[s] + ed[d]);
    atomicMax(mmax + d, enc_f32(e));
}

// Pass 2 (fused): w = exp(e - m[dst]);  denom[dst] += w;  acc[dst,:] += w * H[src,:]
// One wave per edge; lanes carry float2 slices of the 64-wide row (coalesced 256B gather).
__global__ void gat_edge_accum(const int* __restrict__ eidx, int E, int n,
                               const float* __restrict__ es, const float* __restrict__ ed,
                               const int* __restrict__ mmax,
                               float* __restrict__ denom,
                               const float* __restrict__ H, float* __restrict__ acc) {
    int wid  = (blockIdx.x * blockDim.x + threadIdx.x) >> 5;
    int lane = threadIdx.x & 31;
    int total = E + n;
    if (wid >= total) return;
    int s, d;
    if (wid < E) { s = eidx[wid]; d = eidx[E + wid]; } else { s = d = wid - E; }
    float e = lrelu02(es[s] + ed[d]);
    float w = __expf(e - dec_f32(mmax[d]));
    if (lane == 0) atomicAdd(denom + d, w);
    float2 v = ((const float2*)(H + s * 64))[lane];
    float* ap = acc + d * 64 + lane * 2;
    atomicAdd(ap,     v.x * w);
    atomicAdd(ap + 1, v.y * w);
}

// out = acc/denom + bias, optional ReLU, in place
__global__ void gat_bias_act(float* __restrict__ acc, const float* __restrict__ denom,
                             const float* __restrict__ bias, int n, int do_relu) {
    int i = blockIdx.x * blockDim.x + threadIdx.x;
    if (i >= n * 64) return;
    float v = acc[i] / denom[i >> 6] + bias[i & 63];
    if (do_relu) v = v > 0.f ? v : 0.f;
    acc[i] = v;
}

// 2-way branch softmax fusion: fused = a0*Hi + a1*Hp, scores via att[2,64]
__global__ void gat_branch_fuse(const float* __restrict__ Hi, const float* __restrict__ Hp,
                                const float* __restrict__ att, float* __restrict__ fused, int n) {
    int i = blockIdx.x * blockDim.x + threadIdx.x;
    if (i >= n) return;
    const float4* A  = (const float4*)(Hi + i * 64);
    const float4* B  = (const float4*)(Hp + i * 64);
    const float4* w0 = (const float4*)att;
    const float4* w1 = (const float4*)(att + 64);
    float s0 = 0.f, s1 = 0.f;
    #pragma unroll
    for (int j = 0; j < 16; ++j) {
        float4 a = A[j], b = B[j], u = w0[j], v = w1[j];
        s0 += a.x*u.x + a.y*u.y + a.z*u.z + a.w*u.w;
        s1 += b.x*v.x + b.y*v.y + b.z*v.z + b.w*v.w;
    }
    float m  = fmaxf(s0, s1);
    float e0 = __expf(s0 - m), e1 = __expf(s1 - m);
    float inv = 1.f / (e0 + e1);
    float a0 = e0 * inv, a1 = e1 * inv;
    float4* F = (float4*)(fused + i * 64);
    #pragma unroll
    for (int j = 0; j < 16; ++j) {
        float4 a = A[j], b = B[j], r;
        r.x = a0*a.x + a1*b.x; r.y = a0*a.y + a1*b.y;
        r.z = a0*a.z + a1*b.z; r.w = a0*a.w + a1*b.w;
        F[j] = r;
    }
}

// final: branch softmax over (Gi,Gp) with att[2,64], then dot with mlp_w + mlp_b
__global__ void gat_final_mlp(const float* __restrict__ Gi, const float* __restrict__ Gp,
                              const float* __restrict__ att, const float* __restrict__ mlp_w,
                              const float* __restrict__ mlp_b, float* __restrict__ out, int n) {
    int i = blockIdx.x * blockDim.x + threadIdx.x;
    if (i >= n) return;
    const float4* A  = (const float4*)(Gi + i * 64);
    const float4* B  = (const float4*)(Gp + i * 64);
    const float4* w0 = (const float4*)att;
    const float4* w1 = (const float4*)(att + 64);
    const float4* mw = (const float4*)mlp_w;
    float s0 = 0.f, s1 = 0.f;
    #pragma unroll
    for (int j = 0; j < 16; ++j) {
        float4 a = A[j], b = B[j], u = w0[j], v = w1[j];
        s0 += a.x*u.x + a.y*u.y + a.z*u.z + a.w*u.w;
        s1 += b.x*v.x + b.y*v.y + b.z*v.z + b.w*v.w;
    }
    float m  = fmaxf(s0, s1);
    float e0 = __expf(s0 - m), e1 = __expf(s1 - m);
    float inv = 1.f / (e0 + e1);
    float a0 = e0 * inv, a1 = e1 * inv;
    float acc = 0.f;
    #pragma unroll
    for (int j = 0; j < 16; ++j) {
        float4 a = A[j], b = B[j], w = mw[j];
        acc += (a0*a.x + a1*b.x) * w.x + (a0*a.y + a1*b.y) * w.y
             + (a0*a.z + a1*b.z) * w.z + (a0*a.w + a1*b.w) * w.w;
    }
    out[i] = acc + mlp_b[0];
}

// ---------------- host-side conv driver ----------------

static void run_gat_conv(const float* X, const int* eidx, int E, int n,
                         const float* W, const float* asrc, const float* adst,
                         const float* bias, float* hbuf, float* acc,
                         float* es, float* ed, float* denom, int* mmax,
                         int do_relu, hipStream_t stream) {
    int waves   = (n / 16) * 4;
    int gblocks = (waves * 32 + 255) / 256;
    gat_gemm_n64<<<gblocks, 256, 0, stream>>>(X, W, hbuf, n);
    gat_node_dots<<<(n + 255) / 256, 256, 0, stream>>>(hbuf, asrc, adst, es, ed, n);
    gat_init_nd<<<(n + 255) / 256, 256, 0, stream>>>(mmax, denom, n);
    gat_zero_f32<<<(n * 64 + 255) / 256, 256, 0, stream>>>(acc, n * 64);
    int total = E + n;
    gat_edge_max<<<(total + 255) / 256, 256, 0, stream>>>(eidx, E, n, es, ed, mmax);
    long long athreads = (long long)total * 32;
    gat_edge_accum<<<(int)((athreads + 255) / 256), 256, 0, stream>>>(eidx, E, n, es, ed, mmax,
                                                                      denom, hbuf, acc);
    gat_bias_act<<<(n * 64 + 255) / 256, 256, 0, stream>>>(acc, denom, bias, n, do_relu);
}

extern "C" void kernel_launch(void* const* d_in, const int* in_sizes, int n_in,
                              void* d_out, int out_size, void* d_ws, size_t ws_size,
                              hipStream_t stream) {
    const float* x_ind  = (const float*)d_in[0];
    const float* x_pos  = (const float*)d_in[1];
    const int*   e_ind  = (const int*)d_in[2];
    const int*   e_pos  = (const int*)d_in[3];
    const float* w1_ind = (const float*)d_in[4];
    const float* as1_ind= (const float*)d_in[5];
    const float* ad1_ind= (const float*)d_in[6];
    const float* b1_ind = (const float*)d_in[7];
    const float* w2_ind = (const float*)d_in[8];
    const float* as2_ind= (const float*)d_in[9];
    const float* ad2_ind= (const float*)d_in[10];
    const float* b2_ind = (const float*)d_in[11];
    const float* w1_pos = (const float*)d_in[12];
    const float* as1_pos= (const float*)d_in[13];
    const float* ad1_pos= (const float*)d_in[14];
    const float* b1_pos = (const float*)d_in[15];
    const float* w2_pos = (const float*)d_in[16];
    const float* as2_pos= (const float*)d_in[17];
    const float* ad2_pos= (const float*)d_in[18];
    const float* b2_pos = (const float*)d_in[19];
    const float* fh_att = (const float*)d_in[20];
    const float* att_w  = (const float*)d_in[21];
    const float* mlp_w  = (const float*)d_in[22];
    const float* mlp_b  = (const float*)d_in[23];

    int n = in_sizes[0] / 64;      // 50000
    int E = in_sizes[2] / 2;       // 1600000

    // workspace layout
    size_t big = (size_t)n * 64;
    float* hbuf  = (float*)d_ws;        // per-conv h = X@W
    float* b1    = hbuf + big;          // h1_i, later g_i
    float* b2    = b1 + big;            // h1_p, later g_p
    float* fused = b2 + big;            // fused features
    float* es    = fused + big;
    float* ed    = es + n;
    float* denom = ed + n;
    int*   mmax  = (int*)(denom + n);

    // layer 1
    run_gat_conv(x_ind, e_ind, E, n, w1_ind, as1_ind, ad1_ind, b1_ind,
                 hbuf, b1, es, ed, denom, mmax, /*relu=*/1, stream);
    run_gat_conv(x_pos, e_pos, E, n, w1_pos, as1_pos, ad1_pos, b1_pos,
                 hbuf, b2, es, ed, denom, mmax, /*relu=*/1, stream);
    // branch fusion
    gat_branch_fuse<<<(n + 255) / 256, 256, 0, stream>>>(b1, b2, fh_att, fused, n);
    // layer 2 (reuse b1/b2 for g_i/g_p)
    run_gat_conv(fused, e_ind, E, n, w2_ind, as2_ind, ad2_ind, b2_ind,
                 hbuf, b1, es, ed, denom, mmax, /*relu=*/0, stream);
    run_gat_conv(fused, e_pos, E, n, w2_pos, as2_pos, ad2_pos, b2_pos,
                 hbuf, b2, es, ed, denom, mmax, /*relu=*/0, stream);
    // final attention + MLP -> [N,1]
    gat_final_mlp<<<(n + 255) / 256, 256, 0, stream>>>(b1, b2, att_w, mlp_w, mlp_b,
                                                       (float*)d_out, n);
}